// Policy_40123584479691
// MI455X (gfx1250) — compile-verified
//
#include <hip/hip_runtime.h>
#include <stdint.h>

// Problem shape (fixed by the reference)
#define NB 1024   // batch
#define NA 25     // agents
#define NN 250    // graph size
#define NP 250    // patrol nodes

typedef float v4f __attribute__((ext_vector_type(4)));

// Raw 32-bit LDS byte offset of a __shared__ object (generic -> addrspace(3) cast
// folds to the compiler-assigned static LDS offset).
static __device__ __forceinline__ uint32_t lds_addr_of(const void* p) {
    return (uint32_t)(uint64_t)(__attribute__((address_space(3))) const void*)p;
}

__global__ __launch_bounds__(128) void policy_gather_concat_kernel(
    const float* __restrict__ state,    // [B, A, N, 3]
    const float* __restrict__ coords,   // [P, 2]
    const float* __restrict__ adj,      // [P, N]
    const int*   __restrict__ nlv,      // [B, A]
    const int*   __restrict__ pidx,     // [P]
    float*       __restrict__ out)      // [B, A, N, 6]
{
    __shared__ float s_state[NN * 3];   // 750 floats, row of state for this (b,a)
    __shared__ float s_adj[NN + 6];     // 250 floats (+pad), gathered adj row

    const int pair = blockIdx.x;        // flat (b*A + a), 0..25599
    const int t    = threadIdx.x;       // 0..127

    // Block-uniform gathers (scalar path)
    const int   v   = nlv[pair];
    const int   idx = pidx[v];
    const float zb0 = coords[2 * v + 0];
    const float zb1 = coords[2 * v + 1];

    const float* sbase = state + (size_t)pair * (NN * 3);  // 3000 B pitch (8B aligned)
    const float* abase = adj   + (size_t)idx  * NN;        // 1000 B pitch (8B aligned)

    const uint32_t lds_state = lds_addr_of(s_state);
    const uint32_t lds_adj   = lds_addr_of(s_adj);

    // ---- Async global -> LDS staging (CDNA5 async path, tracked by ASYNCcnt) ----
    // state row: 750 floats = 375 x b64 chunks
    for (int i = t; i < (NN * 3) / 2; i += 128) {
        const float*   g = sbase + 2 * i;
        const uint32_t l = lds_state + 8u * (uint32_t)i;
        asm volatile("global_load_async_to_lds_b64 %0, %1, off"
                     :: "v"(l), "v"(g) : "memory");
    }
    // adj row: 250 floats = 125 x b64 chunks
    for (int i = t; i < NN / 2; i += 128) {
        const float*   g = abase + 2 * i;
        const uint32_t l = lds_adj + 8u * (uint32_t)i;
        asm volatile("global_load_async_to_lds_b64 %0, %1, off"
                     :: "v"(l), "v"(g) : "memory");
    }
    // Each wave drains its own async ops, then the workgroup barrier makes LDS
    // globally visible within the block.
    asm volatile("s_wait_asynccnt 0" ::: "memory");
    __syncthreads();

    // ---- Compose + coalesced non-temporal b128 stores ----
    // Output row = 1500 floats = 6000 bytes = 375 float4 chunks; 6000 % 16 == 0,
    // so every block's segment is 16B aligned.
    v4f* outv = (v4f*)(out + (size_t)pair * (NN * 6));
    for (int q = t; q < (NN * 6) / 4; q += 128) {
        float r[4];
#pragma unroll
        for (int j = 0; j < 4; ++j) {
            const int e = 4 * q + j;       // element index in the 1500-float row
            const int n = e / 6;           // node      (magic-multiply division)
            const int c = e - 6 * n;       // channel 0..5
            float x;
            if (c < 3)       x = s_state[3 * n + c];  // state features
            else if (c == 3) x = zb0;                 // zuobiao.x
            else if (c == 4) x = zb1;                 // zuobiao.y
            else             x = s_adj[n];            // mask channel
            r[j] = x;
        }
        const v4f rv = { r[0], r[1], r[2], r[3] };
        // Output is streamed (write-once, never re-read): keep it out of caches.
        __builtin_nontemporal_store(rv, &outv[q]);
    }
}

extern "C" void kernel_launch(void* const* d_in, const int* in_sizes, int n_in,
                              void* d_out, int out_size, void* d_ws, size_t ws_size,
                              hipStream_t stream) {
    (void)in_sizes; (void)n_in; (void)out_size; (void)d_ws; (void)ws_size;

    const float* state  = (const float*)d_in[0];  // [B,A,N,3] f32
    const float* coords = (const float*)d_in[1];  // [P,2]     f32
    const float* adj    = (const float*)d_in[2];  // [P,N]     f32
    const int*   nlv    = (const int*)d_in[3];    // [B,A]     i32
    const int*   pidx   = (const int*)d_in[4];    // [P]       i32
    float*       out    = (float*)d_out;          // [B,A,N,6] f32

    dim3 grid(NB * NA);   // 25,600 blocks, one (batch, agent) pair each
    dim3 block(128);      // 4 wave32 waves
    policy_gather_concat_kernel<<<grid, block, 0, stream>>>(
        state, coords, adj, nlv, pidx, out);
}